// DiceLoss_61203283968424
// MI455X (gfx1250) — compile-verified
//
#include <hip/hip_runtime.h>
#include <hip/hip_bf16.h>

// DiceLoss: pred (8,19,512,512) f32 logits, target (8,512,512) i32 -> scalar f32.
// Bandwidth-bound: 167 MB in / 4 B out => ~7.2 us floor at 23.3 TB/s.
// exp/rcp ride the trans pipe (co-executes with VALU); loads are NT b128 streams.

#define NC        19
#define NB        8
#define HW        (512 * 512)
#define IGNORE_IX 255
#define SMOOTHF   1e-5f
#define BLOCK     256
#define ITERS     4
#define PIX_PER_BLOCK (BLOCK * 4 * ITERS)   // 4096 pixels per block
#define BPB       (HW / PIX_PER_BLOCK)      // 64 blocks per batch image
#define NACC      (3 * NC)                  // 57: [sumP | inter | cnt]

typedef __attribute__((ext_vector_type(2))) float v2f;
typedef __attribute__((ext_vector_type(4))) float v4f;
typedef __attribute__((ext_vector_type(8))) float v8f;
typedef __attribute__((ext_vector_type(4))) int   v4i;

#if __has_builtin(__builtin_amdgcn_wmma_f32_16x16x4_f32)
#define HAVE_WMMA4 1
#endif

__device__ __forceinline__ float fast_rcp(float x) {
#if __has_builtin(__builtin_amdgcn_rcpf)
    return __builtin_amdgcn_rcpf(x);   // v_rcp_f32 (1 ulp, trans pipe)
#else
    return 1.0f / x;
#endif
}

__global__ __launch_bounds__(BLOCK) void dice_partial_kernel(
    const float* __restrict__ pred, const int* __restrict__ target,
    float* __restrict__ ws)
{
    const int b   = blockIdx.x / BPB;
    const int blk = blockIdx.x % BPB;
    const int tid = threadIdx.x;

    const float* predb = pred   + (size_t)b * NC * HW;
    const int*   tgtb  = target + (size_t)b * HW;

    float acc[NACC];
    #pragma unroll
    for (int q = 0; q < NACC; ++q) acc[q] = 0.0f;

    #pragma unroll
    for (int it = 0; it < ITERS; ++it) {
        const int pix = blk * PIX_PER_BLOCK + it * (BLOCK * 4) + tid * 4;

        // 19 coalesced non-temporal b128 streams (one per channel plane)
        float v[NC][4];
        #pragma unroll
        for (int c = 0; c < NC; ++c) {
            const v4f r = __builtin_nontemporal_load(
                reinterpret_cast<const v4f*>(predb + (size_t)c * HW + pix));
            v[c][0] = r[0]; v[c][1] = r[1]; v[c][2] = r[2]; v[c][3] = r[3];
        }
        const v4i t4 = *reinterpret_cast<const v4i*>(tgtb + pix);
        const int tt[4] = { t4[0], t4[1], t4[2], t4[3] };

        #pragma unroll
        for (int j = 0; j < 4; ++j) {
            float m = v[0][j];
            #pragma unroll
            for (int c = 1; c < NC; ++c) m = fmaxf(m, v[c][j]);
            float e[NC];
            float s = 0.0f;
            #pragma unroll
            for (int c = 0; c < NC; ++c) { e[c] = __expf(v[c][j] - m); s += e[c]; }
            const int   t    = tt[j];
            const float keep = (t != IGNORE_IX) ? 1.0f : 0.0f;  // masked pixel => no contribution
            const float inv  = keep * fast_rcp(s);
            #pragma unroll
            for (int c = 0; c < NC; ++c) {
                const float p  = e[c] * inv;
                const float is = (t == c) ? 1.0f : 0.0f;
                acc[c]          += p;       // sum of masked softmax probs
                acc[NC + c]     += p * is;  // intersection
                acc[2 * NC + c] += is;      // one-hot count
            }
        }
    }

    // wave32 shuffle reduction (ds_bpermute path), then cross-wave via LDS
    const int lane = tid & 31;
    const int wv   = tid >> 5;
    __shared__ float red[BLOCK / 32][NACC];
    #pragma unroll
    for (int q = 0; q < NACC; ++q) {
        float x = acc[q];
        #pragma unroll
        for (int off = 16; off > 0; off >>= 1) x += __shfl_down(x, off, 32);
        if (lane == 0) red[wv][q] = x;
    }
    __syncthreads();
    if (tid < NACC) {
        float s = 0.0f;
        #pragma unroll
        for (int w = 0; w < BLOCK / 32; ++w) s += red[w][tid];
        atomicAdd(ws + b * NACC + tid, s);  // global_atomic_add_f32
    }
}

// Single-wave finisher: 152 dice losses -> mean, summed with V_WMMA_F32_16X16X4_F32
// (four 16x4 A-tiles of loss values x all-ones 4x16 B, accumulated in C).
__global__ __launch_bounds__(32) void dice_final_kernel(
    const float* __restrict__ ws, float* __restrict__ out)
{
    __shared__ float vals[256];
    const int lane = threadIdx.x;

    for (int i = lane; i < 256; i += 32) {
        float loss = 0.0f;
        if (i < NB * NC) {
            const int b = i / NC, c = i % NC;
            const float sp = ws[b * NACC + c];
            const float in = ws[b * NACC + NC + c];
            const float cn = ws[b * NACC + 2 * NC + c];
            loss = 1.0f - (2.0f * in + SMOOTHF) / (sp + cn + SMOOTHF);
        }
        vals[i] = loss;
    }
    __syncthreads();

#ifdef HAVE_WMMA4
    // D(m,n) = sum_q sum_k A_q(m,k)  with B = ones(4x16); EXEC is all-ones here.
    v8f c8 = {};
    v2f bones = { 1.0f, 1.0f };
    #pragma unroll
    for (int q = 0; q < 4; ++q) {
        v2f a;
        a[0] = vals[q * 64 + lane * 2 + 0];
        a[1] = vals[q * 64 + lane * 2 + 1];
        c8 = __builtin_amdgcn_wmma_f32_16x16x4_f32(
            false, a, false, bones, (short)0, c8, false, false);
    }
    // lanes 0-15 hold rows M=0..7 (vgpr j -> M=j), lanes 16-31 hold M=8..15
    float half = 0.0f;
    #pragma unroll
    for (int j = 0; j < 8; ++j) half += c8[j];
    const float total = __shfl(half, 0, 32) + __shfl(half, 16, 32);
#else
    float local = 0.0f;
    for (int i = lane; i < 256; i += 32) local += vals[i];
    float total = local;
    #pragma unroll
    for (int off = 16; off > 0; off >>= 1) total += __shfl_down(total, off, 32);
    total = __shfl(total, 0, 32);
#endif

    if (lane == 0) out[0] = total * (1.0f / (float)(NB * NC));
}

extern "C" void kernel_launch(void* const* d_in, const int* in_sizes, int n_in,
                              void* d_out, int out_size, void* d_ws, size_t ws_size,
                              hipStream_t stream) {
    (void)in_sizes; (void)n_in; (void)out_size; (void)ws_size;
    const float* pred   = (const float*)d_in[0];
    const int*   target = (const int*)d_in[1];
    float*       ws     = (float*)d_ws;

    (void)hipMemsetAsync(d_ws, 0, NB * NACC * sizeof(float), stream);
    dice_partial_kernel<<<dim3(NB * BPB), dim3(BLOCK), 0, stream>>>(pred, target, ws);
    dice_final_kernel<<<1, 32, 0, stream>>>(ws, (float*)d_out);
}